// FraudDetectionGNN_2869038154542
// MI455X (gfx1250) — compile-verified
//
#include <hip/hip_runtime.h>
#include <math.h>

typedef float v2f __attribute__((ext_vector_type(2)));
typedef float v8f __attribute__((ext_vector_type(8)));

// ---------------------------------------------------------------------------
// helpers
// ---------------------------------------------------------------------------
__device__ __forceinline__ float lrelu02(float x) { return x > 0.f ? x : 0.2f * x; }

// Float atomic max via sign-split integer atomics (IEEE ordering trick).
// For x >= 0 int ordering matches float ordering; for x < 0 uint ordering is
// reversed, so atomicMin(uint) implements float max. Works for mixed signs.
__device__ __forceinline__ void atomicMaxF(float* addr, float val) {
  if (val >= 0.f) atomicMax((int*)addr, __float_as_int(val));
  else            atomicMin((unsigned int*)addr, __float_as_uint(val));
}

// ---------------------------------------------------------------------------
// GEMM: Y[N,C] = X[N,K] @ W[K,C], f32 via V_WMMA_F32_16X16X4_F32.
// One wave computes a 16x64 output tile; 4 WMMAs per K-step of 4.
// ---------------------------------------------------------------------------
__global__ void gemm_wmma_f32(const float* __restrict__ X, const float* __restrict__ W,
                              float* __restrict__ Y, int N, int K, int C) {
  const int wave = threadIdx.x >> 5;
  const int lane = threadIdx.x & 31;
  const int rowTile = blockIdx.x * (blockDim.x >> 5) + wave;
  if (rowTile * 16 >= N) return;
  const int row0 = rowTile * 16;
  const int col0 = blockIdx.y * 64;
  const int m  = lane & 15;          // A row this lane holds
  const int kk = (lane >> 4) << 1;   // K sub-pair: 0 or 2
  const int nl = lane & 15;          // B/D column this lane holds

  v8f acc0 = {}, acc1 = {}, acc2 = {}, acc3 = {};
  const float* xrow = X + (size_t)(row0 + m) * K;

  for (int k = 0; k < K; k += 4) {
    v2f a;
    a.x = xrow[k + kk];
    a.y = xrow[k + kk + 1];
    const float* w0 = W + (size_t)(k + kk) * C + col0 + nl;       // row k+kk
    const float* w1 = W + (size_t)(k + kk + 1) * C + col0 + nl;   // row k+kk+1
    v2f b0; b0.x = w0[0];  b0.y = w1[0];
    v2f b1; b1.x = w0[16]; b1.y = w1[16];
    v2f b2; b2.x = w0[32]; b2.y = w1[32];
    v2f b3; b3.x = w0[48]; b3.y = w1[48];
    acc0 = __builtin_amdgcn_wmma_f32_16x16x4_f32(false, a, false, b0, (short)0, acc0, false, false);
    acc1 = __builtin_amdgcn_wmma_f32_16x16x4_f32(false, a, false, b1, (short)0, acc1, false, false);
    acc2 = __builtin_amdgcn_wmma_f32_16x16x4_f32(false, a, false, b2, (short)0, acc2, false, false);
    acc3 = __builtin_amdgcn_wmma_f32_16x16x4_f32(false, a, false, b3, (short)0, acc3, false, false);
  }

  const int mo = (lane >> 4) * 8;  // D: vgpr r holds row r (lanes 0-15) / r+8 (lanes 16-31)
#pragma unroll
  for (int r = 0; r < 8; ++r) {
    float* yrow = Y + (size_t)(row0 + mo + r) * C + col0 + nl;
    yrow[0]  = acc0[r];
    yrow[16] = acc1[r];
    yrow[32] = acc2[r];
    yrow[48] = acc3[r];
  }
}

// ---------------------------------------------------------------------------
// Attention scores: a_s[n,h] = sum_c h[n,h,c]*att_s[h,c]  (warp per (n,h))
// ---------------------------------------------------------------------------
__global__ void attn_scores(const float* __restrict__ Hf, const float* __restrict__ att_s,
                            const float* __restrict__ att_d, float* __restrict__ as_,
                            float* __restrict__ ad_, int N, int NH) {
  int warp = (int)((blockIdx.x * (long long)blockDim.x + threadIdx.x) >> 5);
  int lane = threadIdx.x & 31;
  if (warp >= N * NH) return;
  int h = warp % NH;
  const float* hp = Hf + (size_t)warp * 64;
  const float* sp = att_s + h * 64;
  const float* dp = att_d + h * 64;
  float v0 = hp[lane], v1 = hp[lane + 32];
  float s = v0 * sp[lane] + v1 * sp[lane + 32];
  float d = v0 * dp[lane] + v1 * dp[lane + 32];
  for (int off = 16; off; off >>= 1) {
    s += __shfl_down(s, off, 32);
    d += __shfl_down(d, off, 32);
  }
  if (lane == 0) { as_[warp] = s; ad_[warp] = d; }
}

// Self-loop seeds the segment max: m[n,h] = lrelu(a_s[n,h] + a_d[n,h])
__global__ void init_self_max(const float* __restrict__ as_, const float* __restrict__ ad_,
                              float* __restrict__ m, int NHtot) {
  int i = blockIdx.x * blockDim.x + threadIdx.x;
  if (i >= NHtot) return;
  m[i] = lrelu02(as_[i] + ad_[i]);
}

__global__ void edge_max(const int* __restrict__ src, const int* __restrict__ dst,
                         const float* __restrict__ as_, const float* __restrict__ ad_,
                         float* __restrict__ m, int EH, int NH) {
  int t = blockIdx.x * blockDim.x + threadIdx.x;
  if (t >= EH) return;
  int e = t / NH, h = t - e * NH;
  int s = src[e], d = dst[e];
  float ev = lrelu02(as_[s * NH + h] + ad_[d * NH + h]);
  atomicMaxF(&m[d * NH + h], ev);
}

// Self-loop seeds the denom: denom[n,h] = exp(e_self - m[n,h])
__global__ void init_self_denom(const float* __restrict__ as_, const float* __restrict__ ad_,
                                const float* __restrict__ m, float* __restrict__ denom, int NHtot) {
  int i = blockIdx.x * blockDim.x + threadIdx.x;
  if (i >= NHtot) return;
  float ev = lrelu02(as_[i] + ad_[i]);
  denom[i] = __expf(ev - m[i]);
}

__global__ void edge_denom(const int* __restrict__ src, const int* __restrict__ dst,
                           const float* __restrict__ as_, const float* __restrict__ ad_,
                           const float* __restrict__ m, float* __restrict__ denom,
                           int EH, int NH) {
  int t = blockIdx.x * blockDim.x + threadIdx.x;
  if (t >= EH) return;
  int e = t / NH, h = t - e * NH;
  int s = src[e], d = dst[e];
  int di = d * NH + h;
  float ev = lrelu02(as_[s * NH + h] + ad_[di]);
  atomicAdd(&denom[di], __expf(ev - m[di]));
}

// Self-loop message seeds the output: out[n,:] = h[n,:] * alpha_self
__global__ void init_out_self(const float* __restrict__ Hf, const float* __restrict__ as_,
                              const float* __restrict__ ad_, const float* __restrict__ m,
                              const float* __restrict__ denom, float* __restrict__ out,
                              int total /* N*NH*64 */) {
  int i = blockIdx.x * blockDim.x + threadIdx.x;
  if (i >= total) return;
  int nh = i >> 6;  // = n*NH + h (head-major 64-chunks)
  float ev = lrelu02(as_[nh] + ad_[nh]);
  float alpha = __expf(ev - m[nh]) / (denom[nh] + 1e-16f);
  out[i] = Hf[i] * alpha;
}

// Warp-per-edge aggregation. Per-head alpha computed once by lanes < NH and
// broadcast; lanes sweep channels in coalesced 32-strides; f32 global atomics.
__global__ void edge_aggregate(const int* __restrict__ src, const int* __restrict__ dst,
                               const float* __restrict__ Hf, const float* __restrict__ as_,
                               const float* __restrict__ ad_, const float* __restrict__ m,
                               const float* __restrict__ denom, float* __restrict__ out,
                               int E, int NH) {
  int warp = (int)((blockIdx.x * (long long)blockDim.x + threadIdx.x) >> 5);
  int lane = threadIdx.x & 31;
  if (warp >= E) return;
  int s = src[warp], d = dst[warp];
  int HC = NH * 64;
  const float* hs = Hf + (size_t)s * HC;
  __builtin_prefetch(hs, 0, 0);  // global_prefetch_b8 on the gathered row
  float alpha = 0.f;
  if (lane < NH) {
    int di = d * NH + lane;
    float ev = lrelu02(as_[s * NH + lane] + ad_[di]);
    alpha = __expf(ev - m[di]) / (denom[di] + 1e-16f);
  }
  float* od = out + (size_t)d * HC;
#pragma unroll
  for (int j = 0; j < 8; ++j) {  // up to HC=256 channels
    int c = lane + 32 * j;
    if (c >= HC) break;
    float al = __shfl(alpha, c >> 6, 32);
    atomicAdd(&od[c], hs[c] * al);
  }
}

// ---------------------------------------------------------------------------
// BatchNorm stats (block per channel) + fused BN-apply + ELU
// ---------------------------------------------------------------------------
__global__ void bn_stats(const float* __restrict__ X, const float* __restrict__ g,
                         float* __restrict__ mu, float* __restrict__ sc, int N, int C) {
  int c = blockIdx.x;
  float s = 0.f, ss = 0.f;
  for (int n = threadIdx.x; n < N; n += blockDim.x) {
    float v = X[(size_t)n * C + c];
    s += v; ss += v * v;
  }
  __shared__ float sh[512];
  sh[threadIdx.x] = s; sh[threadIdx.x + 256] = ss;
  __syncthreads();
  for (int off = 128; off; off >>= 1) {
    if ((int)threadIdx.x < off) {
      sh[threadIdx.x]       += sh[threadIdx.x + off];
      sh[threadIdx.x + 256] += sh[threadIdx.x + 256 + off];
    }
    __syncthreads();
  }
  if (threadIdx.x == 0) {
    float mean = sh[0] / (float)N;
    float var  = sh[256] / (float)N - mean * mean;
    mu[c] = mean;
    sc[c] = g[c] * rsqrtf(var + 1e-5f);
  }
}

__global__ void bn_elu(const float* __restrict__ X, const float* __restrict__ mu,
                       const float* __restrict__ sc, const float* __restrict__ be,
                       float* __restrict__ Y, int total, int C) {
  int i = blockIdx.x * blockDim.x + threadIdx.x;
  if (i >= total) return;
  int c = i % C;
  float v = (X[i] - mu[c]) * sc[c] + be[c];
  Y[i] = v > 0.f ? v : __expf(v) - 1.f;
}

// ---------------------------------------------------------------------------
// Global mean pool + MLP head
// ---------------------------------------------------------------------------
__global__ void mean_pool(const float* __restrict__ X, float* __restrict__ pooled, int N, int C) {
  int c = blockIdx.x;
  float s = 0.f;
  for (int n = threadIdx.x; n < N; n += blockDim.x) s += X[(size_t)n * C + c];
  __shared__ float sh[256];
  sh[threadIdx.x] = s;
  __syncthreads();
  for (int off = 128; off; off >>= 1) {
    if ((int)threadIdx.x < off) sh[threadIdx.x] += sh[threadIdx.x + off];
    __syncthreads();
  }
  if (threadIdx.x == 0) pooled[c] = sh[0] / (float)N;
}

__global__ void mlp_head(const float* __restrict__ pooled, const float* __restrict__ fw1,
                         const float* __restrict__ fb1, const float* __restrict__ fw2,
                         const float* __restrict__ fb2, float* __restrict__ out) {
  __shared__ float p[64], r1[32];
  int t = threadIdx.x;
  if (t < 64) p[t] = pooled[t];
  __syncthreads();
  if (t < 32) {
    float s = fb1[t];
    for (int c = 0; c < 64; ++c) s += p[c] * fw1[c * 32 + t];
    r1[t] = s > 0.f ? s : 0.f;
  }
  __syncthreads();
  if (t < 2) {
    float o = fb2[t];
    for (int j = 0; j < 32; ++j) o += r1[j] * fw2[j * 2 + t];
    out[t] = o;
  }
}

// ---------------------------------------------------------------------------
// One full GAT layer (GEMM -> scores -> softmax passes -> aggregate -> BN+ELU)
// Note: the GATConv bias b cancels exactly inside BatchNorm (and is zero in
// setup), so it is skipped.
// ---------------------------------------------------------------------------
static void run_gat_layer(const float* Xin, int Kin, const float* Wmat,
                          const float* att_s, const float* att_d,
                          const float* g, const float* be,
                          const int* src, const int* dst,
                          float* Hbuf, float* Obuf, float* Ybuf,
                          float* as_, float* ad_, float* m, float* denom,
                          float* mu, float* sc,
                          int N, int E, int NH, hipStream_t stream) {
  const int C = NH * 64;
  // 1) feature transform on the matrix pipe
  dim3 gg((unsigned)((N / 16 + 7) / 8), (unsigned)(C / 64));
  gemm_wmma_f32<<<gg, 256, 0, stream>>>(Xin, Wmat, Hbuf, N, Kin, C);
  // 2) per-node attention scores
  long long sthreads = (long long)N * NH * 32;
  attn_scores<<<(unsigned)((sthreads + 255) / 256), 256, 0, stream>>>(Hbuf, att_s, att_d, as_, ad_, N, NH);
  // 3) segment max (self-loop seed + edge atomics)
  int NHtot = N * NH;
  init_self_max<<<(NHtot + 255) / 256, 256, 0, stream>>>(as_, ad_, m, NHtot);
  int EH = E * NH;
  edge_max<<<(EH + 255) / 256, 256, 0, stream>>>(src, dst, as_, ad_, m, EH, NH);
  // 4) segment sum of exp(e - m)
  init_self_denom<<<(NHtot + 255) / 256, 256, 0, stream>>>(as_, ad_, m, denom, NHtot);
  edge_denom<<<(EH + 255) / 256, 256, 0, stream>>>(src, dst, as_, ad_, m, denom, EH, NH);
  // 5) weighted message aggregation
  int total = N * C;
  init_out_self<<<(total + 255) / 256, 256, 0, stream>>>(Hbuf, as_, ad_, m, denom, Obuf, total);
  long long athreads = (long long)E * 32;
  edge_aggregate<<<(unsigned)((athreads + 255) / 256), 256, 0, stream>>>(src, dst, Hbuf, as_, ad_, m, denom, Obuf, E, NH);
  // 6) BatchNorm + ELU
  bn_stats<<<C, 256, 0, stream>>>(Obuf, g, mu, sc, N, C);
  bn_elu<<<(total + 255) / 256, 256, 0, stream>>>(Obuf, mu, sc, be, Ybuf, total, C);
}

extern "C" void kernel_launch(void* const* d_in, const int* in_sizes, int n_in,
                              void* d_out, int out_size, void* d_ws, size_t ws_size,
                              hipStream_t stream) {
  const float* x      = (const float*)d_in[0];
  const int*   ei     = (const int*)d_in[1];
  const float* W1     = (const float*)d_in[2];
  const float* att_s1 = (const float*)d_in[3];
  const float* att_d1 = (const float*)d_in[4];
  const float* W2     = (const float*)d_in[6];
  const float* att_s2 = (const float*)d_in[7];
  const float* att_d2 = (const float*)d_in[8];
  const float* W3     = (const float*)d_in[10];
  const float* att_s3 = (const float*)d_in[11];
  const float* att_d3 = (const float*)d_in[12];
  const float* g1 = (const float*)d_in[14];
  const float* be1 = (const float*)d_in[15];
  const float* g2 = (const float*)d_in[16];
  const float* be2 = (const float*)d_in[17];
  const float* g3 = (const float*)d_in[18];
  const float* be3 = (const float*)d_in[19];
  const float* fw1 = (const float*)d_in[20];
  const float* fb1 = (const float*)d_in[21];
  const float* fw2 = (const float*)d_in[22];
  const float* fb2 = (const float*)d_in[23];

  const int N = in_sizes[0] / 128;
  const int E = in_sizes[1] / 2;
  const int* src = ei;
  const int* dst = ei + E;

  // workspace layout (floats)
  float* ws = (float*)d_ws;
  size_t off = 0;
  float* bufA  = ws + off; off += (size_t)N * 256;
  float* bufB  = ws + off; off += (size_t)N * 256;
  float* as_   = ws + off; off += (size_t)N * 4;
  float* ad_   = ws + off; off += (size_t)N * 4;
  float* mbuf  = ws + off; off += (size_t)N * 4;
  float* denom = ws + off; off += (size_t)N * 4;
  float* mu    = ws + off; off += 256;
  float* sc    = ws + off; off += 256;
  float* pooled = ws + off; off += 64;

  // Layer 1: x[N,128] -> bufA (h1) -> aggregate bufB -> BN+ELU -> bufA
  run_gat_layer(x, 128, W1, att_s1, att_d1, g1, be1, src, dst,
                bufA, bufB, bufA, as_, ad_, mbuf, denom, mu, sc, N, E, 4, stream);
  // Layer 2: bufA[N,256] -> bufB (h2) -> aggregate bufA -> BN+ELU -> bufB
  run_gat_layer(bufA, 256, W2, att_s2, att_d2, g2, be2, src, dst,
                bufB, bufA, bufB, as_, ad_, mbuf, denom, mu, sc, N, E, 4, stream);
  // Layer 3: bufB[N,256] -> bufA (h3,[N,64]) -> aggregate bufB -> BN+ELU -> bufA
  run_gat_layer(bufB, 256, W3, att_s3, att_d3, g3, be3, src, dst,
                bufA, bufB, bufA, as_, ad_, mbuf, denom, mu, sc, N, E, 1, stream);

  // mean pool + MLP head
  mean_pool<<<64, 256, 0, stream>>>(bufA, pooled, N, 64);
  mlp_head<<<1, 64, 0, stream>>>(pooled, fw1, fb1, fw2, fb2, (float*)d_out);
}